// DoppiaBuca_403726926523
// MI455X (gfx1250) — compile-verified
//
#include <hip/hip_runtime.h>

typedef float v2f __attribute__((ext_vector_type(2)));
typedef float v8f __attribute__((ext_vector_type(8)));

#define NDIM   256
#define AUGC   512
#define A2     1.96f      // a^2, a = 1.4
#define DTC    0.1f
#define LIMC   14.0f      // 10*a
#define BETAC  0.0625f    // 1/sqrt(256)

// ---------------------------------------------------------------------------
// 1) Build augmented matrix [base | I] and clipped eigenvalues
// ---------------------------------------------------------------------------
__global__ void setup_aug_eig(const float* __restrict__ bt, const float* __restrict__ bfx,
                              const float* __restrict__ autov, const float* __restrict__ gamma,
                              float* __restrict__ aug, float* __restrict__ eig, int NK) {
  int idx = blockIdx.x * blockDim.x + threadIdx.x;     // 0 .. 256*512-1
  int i = idx >> 9, j = idx & 511;
  float v;
  if (j < NDIM) {
    v = (j < NK) ? bt[i * NK + j] : bfx[i * (NDIM - NK) + (j - NK)];
  } else {
    v = (i == (j - NDIM)) ? 1.0f : 0.0f;
  }
  aug[i * AUGC + j] = v;
  if (idx < NDIM) {
    float g = gamma[0];
    float xmax = 8.0f * A2 * g * 16.0f - 0.1f;         // 8*a^2*gamma*sqrt(N) - 0.1
    float e = (idx < NK) ? autov[idx] : 0.0f;
    e = fminf(fmaxf(e, -1e9f), xmax);
    eig[idx] = e;
  }
}

// ---------------------------------------------------------------------------
// 2) Gauss-Jordan with partial pivoting on [base | I]  (single workgroup)
//    On exit aug[:, 256:512] = inv(base)
// ---------------------------------------------------------------------------
__global__ void __launch_bounds__(1024) gauss_jordan(float* __restrict__ aug) {
  __shared__ float prow[AUGC];
  __shared__ float fac[NDIM];
  __shared__ float redv[NDIM];
  __shared__ int   redi[NDIM];
  const int t = threadIdx.x;
  for (int k = 0; k < NDIM; ++k) {
    // --- partial pivot: argmax |aug[r][k]| over r >= k ---
    if (t < NDIM) { redv[t] = (t >= k) ? fabsf(aug[t * AUGC + k]) : -1.0f; redi[t] = t; }
    __syncthreads();
    for (int s = 128; s > 0; s >>= 1) {
      if (t < s && redv[t + s] > redv[t]) { redv[t] = redv[t + s]; redi[t] = redi[t + s]; }
      __syncthreads();
    }
    const int p = redi[0];
    const float pv = aug[p * AUGC + k];
    __syncthreads();                                   // pv read by all before row writes
    // --- swap rows k<->p, normalize pivot row, stage it in LDS ---
    for (int j = t; j < AUGC; j += 1024) {
      float ak = aug[k * AUGC + j];
      float ap = aug[p * AUGC + j];
      float npj = ap / pv;
      aug[k * AUGC + j] = npj;
      if (p != k) aug[p * AUGC + j] = ak;
      prow[j] = npj;
    }
    __syncthreads();
    if (t < NDIM) fac[t] = (t == k) ? 0.0f : aug[t * AUGC + k];
    __syncthreads();
    // --- eliminate column k from all other rows ---
    for (int e = t; e < NDIM * AUGC; e += 1024) {
      int r = e >> 9, j = e & 511;
      aug[e] = aug[e] - fac[r] * prow[j];
    }
    __syncthreads();
  }
}

// ---------------------------------------------------------------------------
// 3) A = (base * eig[None,:]) @ inv(base)   (tiny 256^3 GEMM)
// ---------------------------------------------------------------------------
__global__ void build_A(const float* __restrict__ bt, const float* __restrict__ bfx,
                        const float* __restrict__ aug, const float* __restrict__ eig,
                        float* __restrict__ Amat, int NK) {
  int i = blockIdx.x, j = threadIdx.x;
  float s = 0.0f;
  for (int l = 0; l < NDIM; ++l) {
    float b = (l < NK) ? bt[i * NK + l] : bfx[i * (NDIM - NK) + (l - NK)];
    s += (b * eig[l]) * aug[l * AUGC + NDIM + j];
  }
  Amat[i * NDIM + j] = s;
}

// ---------------------------------------------------------------------------
// 4) One Euler step:  Y = clip(X + dt*(4|g|(a^2 X - X^3) + (X @ A^T)*beta))
//    8 waves / block; each wave: 32x64 output via fp32 WMMA (16x16x4)
// ---------------------------------------------------------------------------
__global__ void __launch_bounds__(256) dw_step(const float* __restrict__ X,
                                               const float* __restrict__ Amat,
                                               const float* __restrict__ gamma,
                                               float* __restrict__ Y) {
  const int lane = threadIdx.x & 31;
  const int wave = threadIdx.x >> 5;
  const int lm = lane & 15;          // fragment row/col index
  const int lg = lane >> 4;          // lane group (K half)
  const int rowBase = blockIdx.x * 64 + (wave >> 2) * 32;
  const int colBase = (wave & 3) * 64;

  v8f acc[2][4] = {};                // 32x64 fp32 accumulators

  const float* xA = X    + (size_t)(rowBase + lm) * NDIM + 2 * lg; // A-frag base
  const float* bB = Amat + (size_t)(colBase + lm) * NDIM + 2 * lg; // B-frag base (A row-major = A^T col)

#pragma unroll 4
  for (int k0 = 0; k0 < NDIM; k0 += 4) {
    v2f af[2], bfr[4];
#pragma unroll
    for (int rt = 0; rt < 2; ++rt)
      af[rt] = *(const v2f*)(xA + (size_t)rt * 16 * NDIM + k0);
#pragma unroll
    for (int ct = 0; ct < 4; ++ct)
      bfr[ct] = *(const v2f*)(bB + (size_t)ct * 16 * NDIM + k0);
#pragma unroll
    for (int rt = 0; rt < 2; ++rt)
#pragma unroll
      for (int ct = 0; ct < 4; ++ct)
        acc[rt][ct] = __builtin_amdgcn_wmma_f32_16x16x4_f32(
            false, af[rt], false, bfr[ct], (short)0, acc[rt][ct], false, false);
  }

  const float g4 = 4.0f * fabsf(gamma[0]);
#pragma unroll
  for (int rt = 0; rt < 2; ++rt) {
#pragma unroll
    for (int ct = 0; ct < 4; ++ct) {
#pragma unroll
      for (int v = 0; v < 8; ++v) {
        int row = rowBase + rt * 16 + v + 8 * lg;      // C/D layout: M = v + 8*(lane/16)
        int col = colBase + ct * 16 + lm;              //            N = lane%16
        float xo = X[(size_t)row * NDIM + col];
        float dw = g4 * (xo * A2 - xo * xo * xo);
        float xn = xo + DTC * (dw + acc[rt][ct][v] * BETAC);
        xn = fminf(fmaxf(xn, -LIMC), LIMC);
        Y[(size_t)row * NDIM + col] = xn;
      }
    }
  }
}

// ---------------------------------------------------------------------------
extern "C" void kernel_launch(void* const* d_in, const int* in_sizes, int n_in,
                              void* d_out, int out_size, void* d_ws, size_t ws_size,
                              hipStream_t stream) {
  const float* x     = (const float*)d_in[0];
  const float* bt    = (const float*)d_in[1];
  const float* bfx   = (const float*)d_in[2];
  const float* autov = (const float*)d_in[3];
  const float* gamma = (const float*)d_in[4];
  const int NK    = in_sizes[3];              // 246 trainable eigenvalues
  const int Brows = in_sizes[0] / NDIM;       // 16384

  float* ws0  = (float*)d_ws;                 // ping buffer [Brows, 256]
  float* aug  = ws0 + (size_t)Brows * NDIM;   // 256 x 512 augmented matrix
  float* eig  = aug + NDIM * AUGC;            // 256
  float* Amat = eig + NDIM;                   // 256 x 256
  float* out  = (float*)d_out;

  setup_aug_eig<<<(NDIM * AUGC) / 256, 256, 0, stream>>>(bt, bfx, autov, gamma, aug, eig, NK);
  gauss_jordan<<<1, 1024, 0, stream>>>(aug);
  build_A<<<NDIM, NDIM, 0, stream>>>(bt, bfx, aug, eig, Amat, NK);

  const int TMAX = 64;                        // fixed by the reference setup
  const float* src = x;
  for (int s = 0; s < TMAX; ++s) {
    float* dst = (s & 1) ? out : ws0;         // s=63 (odd) -> final in d_out
    dw_step<<<Brows / 64, 256, 0, stream>>>(src, Amat, gamma, dst);
    src = dst;
  }
}